// BayesianAdapter_58248346468824
// MI455X (gfx1250) — compile-verified
//
#include <hip/hip_runtime.h>
#include <hip/hip_bf16.h>

// Problem constants (from reference): D_IN = D_OUT = 4096, RANK = 8, S = 4, N = 8192
// Collapsed rank C = S*RANK = 32.

typedef __attribute__((ext_vector_type(2))) float v2f;
typedef __attribute__((ext_vector_type(8))) float v8f;

#define DIN   4096
#define DOUT  4096
#define NROW  8192
#define CRANK 32
#define KSPLIT 4
#define KCHUNK (DIN / KSPLIT)   // 1024

// ---------------------------------------------------------------------------
// Kernel 0: build UhatT [32][4096] (K-contiguous for WMMA B operand) and
// Vhat [4096][32] (row-major, K-contiguous rows). Folds tau/lambda reparam
// and the 1/S average into Vhat.
// grid: 512 x 256 threads covering 4096*32 elements.
// ---------------------------------------------------------------------------
__global__ void prep_kernel(const float* __restrict__ U_mean,
                            const float* __restrict__ U_logvar,
                            const float* __restrict__ V_mean,
                            const float* __restrict__ V_logvar,
                            const float* __restrict__ tau_mean,
                            const float* __restrict__ tau_logvar,
                            const float* __restrict__ lambda_mean,
                            const float* __restrict__ lambda_logvar,
                            const float* __restrict__ eps_tau,
                            const float* __restrict__ eps_lambda,
                            const float* __restrict__ eps_U,
                            const float* __restrict__ eps_V,
                            float* __restrict__ UhatT,
                            float* __restrict__ Vhat)
{
    int idx = blockIdx.x * blockDim.x + threadIdx.x;   // 0 .. 4096*32-1
    int d = idx >> 5;
    int c = idx & 31;       // c = s*8 + r
    int s = c >> 3;
    int r = c & 7;

    float tau = tau_mean[0] + __expf(0.5f * tau_logvar[0]) * eps_tau[s];
    float lam = lambda_mean[r] + __expf(0.5f * lambda_logvar[r]) * eps_lambda[s * 8 + r];
    float eff = tau * lam;

    int dr  = d * 8 + r;                 // [D, R] row-major
    int sdr = (s * DIN + d) * 8 + r;     // [S, D, R]

    float u = U_mean[dr] + __expf(0.5f * U_logvar[dr]) * eff * eps_U[sdr];
    float v = V_mean[dr] + __expf(0.5f * V_logvar[dr]) * eff * eps_V[sdr];

    UhatT[c * DIN + d]  = u;             // transposed: row c is K(=d)-contiguous
    Vhat[d * CRANK + c] = 0.25f * v;     // 1/S folded in
}

// ---------------------------------------------------------------------------
// Kernel 1: h_part[p] = x[:, k0:k0+1024] @ Uhat[k0:k0+1024, :]
// One wave per (16-row tile, K-chunk). Two 16x16 accumulators cover all 32
// collapsed-rank columns, reusing each A (x) load for both WMMAs.
// grid: (512, 4), block: 32 (one wave; EXEC all ones as WMMA requires).
// ---------------------------------------------------------------------------
__global__ void __launch_bounds__(32)
gemm1_kernel(const float* __restrict__ x,
             const float* __restrict__ UhatT,
             float* __restrict__ hpart)
{
    const int lane = threadIdx.x;
    const int half = lane >> 4;     // 0: K pair {0,1}, 1: K pair {2,3}
    const int lm   = lane & 15;
    const int n0   = blockIdx.x * 16;
    const int k0   = blockIdx.y * KCHUNK;

    // A: lane holds row n0+lm of x; per K-step loads floats {k+2h, k+2h+1}
    const float* xrow = x + (size_t)(n0 + lm) * DIN + 2 * half;
    // B: lane holds column lm (tile0) / 16+lm (tile1) of Uhat == row of UhatT
    const float* b0p = UhatT + (size_t)lm * DIN + 2 * half;
    const float* b1p = UhatT + (size_t)(16 + lm) * DIN + 2 * half;

    v8f acc0 = {};
    v8f acc1 = {};

#pragma unroll 8
    for (int k = k0; k < k0 + KCHUNK; k += 4) {
        v2f a  = *(const v2f*)(xrow + k);
        v2f b0 = *(const v2f*)(b0p + k);
        v2f b1 = *(const v2f*)(b1p + k);
        acc0 = __builtin_amdgcn_wmma_f32_16x16x4_f32(false, a, false, b0,
                                                     (short)0, acc0, false, false);
        acc1 = __builtin_amdgcn_wmma_f32_16x16x4_f32(false, a, false, b1,
                                                     (short)0, acc1, false, false);
    }

    // C/D layout: VGPR j -> row n0 + j + 8*half, col (tile*16 + lm)
    float* hp = hpart + ((size_t)blockIdx.y * NROW + n0 + 8 * half) * CRANK + lm;
#pragma unroll
    for (int j = 0; j < 8; ++j) {
        hp[j * CRANK]      = acc0[j];
        hp[j * CRANK + 16] = acc1[j];
    }
}

// ---------------------------------------------------------------------------
// Kernel 2: out = (sum_p h_part[p]) @ Vhat^T   (K = 32, 8 WMMA steps)
// One wave computes a 16x128 output strip: A (h rows) loaded once (summing
// the 4 split-K partials, all L2-resident), reused across 8 column tiles.
// grid: (512, 32), block: 32.
// ---------------------------------------------------------------------------
__global__ void __launch_bounds__(32)
gemm2_kernel(const float* __restrict__ hpart,
             const float* __restrict__ Vhat,
             float* __restrict__ out)
{
    const int lane = threadIdx.x;
    const int half = lane >> 4;
    const int lm   = lane & 15;
    const int n0   = blockIdx.x * 16;
    const int o0   = blockIdx.y * 128;

    // A operand for each of the 8 K-steps: lane holds row n0+lm,
    // floats {4*kk + 2*half, +1}; sum the KSPLIT partials.
    v2f A[8];
#pragma unroll
    for (int kk = 0; kk < 8; ++kk) {
        const int cofs = 4 * kk + 2 * half;
        v2f s = {};
#pragma unroll
        for (int p = 0; p < KSPLIT; ++p)
            s += *(const v2f*)(hpart + ((size_t)p * NROW + n0 + lm) * CRANK + cofs);
        A[kk] = s;
    }

    v8f acc[8];
    v8f zero = {};
#pragma unroll
    for (int t = 0; t < 8; ++t) acc[t] = zero;

#pragma unroll
    for (int kk = 0; kk < 8; ++kk) {
#pragma unroll
        for (int t = 0; t < 8; ++t) {
            // B: lane holds column o0 + t*16 + lm of Vhat^T == row of Vhat
            v2f b = *(const v2f*)(Vhat + (size_t)(o0 + t * 16 + lm) * CRANK
                                  + 4 * kk + 2 * half);
            acc[t] = __builtin_amdgcn_wmma_f32_16x16x4_f32(false, A[kk], false, b,
                                                           (short)0, acc[t],
                                                           false, false);
        }
    }

#pragma unroll
    for (int t = 0; t < 8; ++t) {
        float* op = out + (size_t)(n0 + 8 * half) * DOUT + o0 + t * 16 + lm;
#pragma unroll
        for (int j = 0; j < 8; ++j)
            op[(size_t)j * DOUT] = acc[t][j];
    }
}

// ---------------------------------------------------------------------------
// Launch. Workspace layout (floats):
//   [0, 131072)          UhatT  [32][4096]         (512 KB)
//   [131072, 262144)     Vhat   [4096][32]         (512 KB)
//   [262144, 1310720)    hpart  [4][8192][32]      (4 MB)
// Total 5 MB.
// ---------------------------------------------------------------------------
extern "C" void kernel_launch(void* const* d_in, const int* in_sizes, int n_in,
                              void* d_out, int out_size, void* d_ws, size_t ws_size,
                              hipStream_t stream)
{
    const float* x             = (const float*)d_in[0];
    const float* U_mean        = (const float*)d_in[1];
    const float* U_logvar      = (const float*)d_in[2];
    const float* V_mean        = (const float*)d_in[3];
    const float* V_logvar      = (const float*)d_in[4];
    const float* tau_mean      = (const float*)d_in[5];
    const float* tau_logvar    = (const float*)d_in[6];
    const float* lambda_mean   = (const float*)d_in[7];
    const float* lambda_logvar = (const float*)d_in[8];
    const float* eps_tau       = (const float*)d_in[9];
    const float* eps_lambda    = (const float*)d_in[10];
    const float* eps_U         = (const float*)d_in[11];
    const float* eps_V         = (const float*)d_in[12];

    float* ws    = (float*)d_ws;
    float* UhatT = ws;
    float* Vhat  = ws + CRANK * DIN;
    float* hpart = ws + 2 * CRANK * DIN;

    prep_kernel<<<(DIN * CRANK) / 256, 256, 0, stream>>>(
        U_mean, U_logvar, V_mean, V_logvar, tau_mean, tau_logvar,
        lambda_mean, lambda_logvar, eps_tau, eps_lambda, eps_U, eps_V,
        UhatT, Vhat);

    gemm1_kernel<<<dim3(NROW / 16, KSPLIT), 32, 0, stream>>>(x, UhatT, hpart);

    gemm2_kernel<<<dim3(NROW / 16, DOUT / 128), 32, 0, stream>>>(
        hpart, Vhat, (float*)d_out);
}